// RedundancyLoss_13142599926418
// MI455X (gfx1250) — compile-verified
//
#include <hip/hip_runtime.h>

// ---------------- workspace layout (floats) ----------------
// [0]            : CE scalar
// [64 .. 64+4096): M  (64x64 pair maxima, strict-upper populated, rest 0)
// [8192 .. )     : F  = relu(fm), transposed to [c][b*3136+hw], 64*25088
// [OFF_S .. )    : S  = 3x3 boxsum(relu(fm)), same layout
#define OFF_CE 0
#define OFF_M  64
#define OFF_F  8192
#define OFF_S  (8192 + 64 * 25088)
// total ~12.9 MB of float workspace

#define AS1 __attribute__((address_space(1)))
#define AS3 __attribute__((address_space(3)))

typedef int v4i_vs __attribute__((vector_size(16)));   // matches builtin param type

#if __has_builtin(__builtin_amdgcn_global_load_async_to_lds_b128)
#define HAS_ASYNC_B128 1
#else
#define HAS_ASYNC_B128 0
#endif

__device__ __forceinline__ void async_copy_b128(const float* gsrc, float* lds_dst) {
#if HAS_ASYNC_B128
  __builtin_amdgcn_global_load_async_to_lds_b128(
      (AS1 v4i_vs*)gsrc, (AS3 v4i_vs*)lds_dst, 0, 0);
#else
  float4 v = *(const float4*)gsrc;
  *(float4*)lds_dst = v;
#endif
}

__device__ __forceinline__ void async_wait_all() {
#if HAS_ASYNC_B128
 #if __has_builtin(__builtin_amdgcn_s_wait_asynccnt)
  __builtin_amdgcn_s_wait_asynccnt(0);
 #else
  asm volatile("s_wait_asynccnt 0" ::: "memory");
 #endif
#endif
}

// ---------------- kernel 1: CE (8x1000 log-softmax) + zero-init M ----------
__global__ void ce_init_kernel(const int* __restrict__ y,
                               const float* __restrict__ yp,
                               float* __restrict__ ws) {
  const int tid = threadIdx.x;
  float* Mws = ws + OFF_M;
  for (int i = tid; i < 4096; i += 256) Mws[i] = 0.f;

  __shared__ float ce_sh[8];
  const int wv = tid >> 5;     // wave (0..7) = row of y_pred
  const int lane = tid & 31;   // wave32
  const float* row = yp + wv * 1000;

  float mx = -3.4e38f;
  for (int j = lane; j < 1000; j += 32) mx = fmaxf(mx, row[j]);
  for (int o = 16; o > 0; o >>= 1) mx = fmaxf(mx, __shfl_xor(mx, o, 32));

  float s = 0.f;
  for (int j = lane; j < 1000; j += 32) s += expf(row[j] - mx);
  for (int o = 16; o > 0; o >>= 1) s += __shfl_xor(s, o, 32);

  if (lane == 0) {
    const int t = y[wv];
    ce_sh[wv] = -(row[t] - mx - logf(s));
  }
  __syncthreads();
  if (tid == 0) {
    float acc = 0.f;
    for (int r = 0; r < 8; ++r) acc += ce_sh[r];
    ws[OFF_CE] = acc * 0.125f;
  }
}

// ---------------- kernel 2: relu + 3x3 boxsum per 56x56 plane -------------
// One block per (b,c) plane; async-stage the raw plane into LDS.
__global__ void relu_boxsum_kernel(const float* __restrict__ fm,
                                   float* __restrict__ fws,
                                   float* __restrict__ sws) {
  __shared__ __attribute__((aligned(16))) float plane[3136];
  const int pid = blockIdx.x;      // b*64 + c  (fm is contiguous per plane)
  const int tid = threadIdx.x;
  const float* src = fm + pid * 3136;

  for (int q = tid; q < 784; q += 256)           // 784 float4 per plane
    async_copy_b128(src + q * 4, plane + q * 4);
  async_wait_all();
  __syncthreads();

  const int b = pid >> 6, c = pid & 63;
  float* fdst = fws + c * 25088 + b * 3136;
  float* sdst = sws + c * 25088 + b * 3136;

  for (int hw = tid; hw < 3136; hw += 256) {
    const int h = hw / 56, w = hw - h * 56;
    float s = 0.f;
#pragma unroll
    for (int dh = -1; dh <= 1; ++dh) {
      const int hh = h + dh;
      if (hh < 0 || hh >= 56) continue;
#pragma unroll
      for (int dw = -1; dw <= 1; ++dw) {
        const int ww = w + dw;
        if (ww < 0 || ww >= 56) continue;
        s += fmaxf(plane[hh * 56 + ww], 0.f);
      }
    }
    fdst[hw] = fmaxf(plane[hw], 0.f);
    sdst[hw] = s;
  }
}

// ---------------- kernel 3: pair-max over pixels ---------------------------
// grid.x = 10 upper-tri 16x16 channel tiles, grid.y = 64 pixel chunks of 392.
// LDS row stride 396 floats: 396 mod 64 = 12 -> 16 distinct banks across the
// 16 f-rows read per cycle; 396*4 bytes is 16B-aligned for B128 async loads.
#define CP   392
#define CPST 396
__global__ void pairmax_kernel(const float* __restrict__ fws,
                               const float* __restrict__ sws,
                               float* __restrict__ Mws) {
  __shared__ __attribute__((aligned(16))) float lds[32 * CPST];
  static const int IT[10] = {0,0,0,0,1,1,1,2,2,3};
  static const int JT[10] = {0,1,2,3,1,2,3,2,3,3};
  const int it = IT[blockIdx.x], jt = JT[blockIdx.x];
  const int chunk = blockIdx.y;
  const int tid = threadIdx.x;

  // stage 16 f-rows (channels it*16..+15) and 16 S-rows (jt*16..+15)
  for (int idx = tid; idx < 32 * 98; idx += 256) {   // 98 float4 per row
    const int row = idx / 98, q = idx - row * 98;
    const float* src = (row < 16)
        ? fws + (it * 16 + row) * 25088 + chunk * CP + q * 4
        : sws + (jt * 16 + (row - 16)) * 25088 + chunk * CP + q * 4;
    async_copy_b128(src, lds + row * CPST + q * 4);
  }
  async_wait_all();
  __syncthreads();

  const int ti = tid & 15, tj = tid >> 4;        // one thread per (i,j) pair
  const float* fr = lds + ti * CPST;
  const float* sr = lds + (16 + tj) * CPST;
  float m = 0.f;
#pragma unroll 4
  for (int p = 0; p < CP; ++p) {
    const float prod = fr[p] * sr[p];
    m = fmaxf(m, prod * prod);
  }

  const int ig = it * 16 + ti, jg = jt * 16 + tj;
  if (ig < jg) {   // strict upper triangle; values are >=0 so uint max == fp max
    const float v = m * (1.0f / 81.0f);
    atomicMax((unsigned int*)(Mws + ig * 64 + jg), __float_as_uint(v));
  }
}

// ---------------- kernel 4: WMMA triangular sum + CE -----------------------
// sum(M) via D = A x ones + C with V_WMMA_F32_16X16X4_F32.
// A 16x4 f32 layout: lanes 0-15 hold K=0,1 (vgpr0,1); lanes 16-31 hold K=2,3.
typedef float v2f __attribute__((ext_vector_type(2)));
typedef float v8f __attribute__((ext_vector_type(8)));

__global__ void finalize_kernel(const float* __restrict__ ws,
                                float* __restrict__ out) {
  const float* M = ws + OFF_M;
  const int lane = threadIdx.x;            // exactly one wave32, EXEC all ones
#if __has_builtin(__builtin_amdgcn_wmma_f32_16x16x4_f32)
  const int row = lane & 15;
  const int khalf = lane >> 4;             // 0 -> K=0,1 ; 1 -> K=2,3
  v8f acc = {};
  for (int rt = 0; rt < 4; ++rt) {         // 4 row-tiles of M accumulate together
    const float* base = M + (rt * 16 + row) * 64;
    for (int ks = 0; ks < 16; ++ks) {      // K covers all 64 columns, 4 at a time
      const int k0 = ks * 4 + khalf * 2;
      v2f a; a.x = base[k0]; a.y = base[k0 + 1];
      v2f b; b.x = 1.0f; b.y = 1.0f;       // ones matrix: layout-invariant
      acc = __builtin_amdgcn_wmma_f32_16x16x4_f32(
          false, a, false, b, (short)0, acc, false, false);
    }
  }
  // D[i,n] = rowsum_i (identical across n). Rows 0-7 live in lanes 0-15,
  // rows 8-15 in lanes 16-31, one per VGPR.
  float t = 0.f;
#pragma unroll
  for (int r = 0; r < 8; ++r) t += acc[r];
  const float total = __shfl(t, 0, 32) + __shfl(t, 16, 32);
  if (lane == 0) out[0] = ws[OFF_CE] + total;
#else
  if (lane == 0) {
    float total = 0.f;
    for (int i = 0; i < 4096; ++i) total += M[i];
    out[0] = ws[OFF_CE] + total;
  }
#endif
}

// ---------------- launch ---------------------------------------------------
extern "C" void kernel_launch(void* const* d_in, const int* in_sizes, int n_in,
                              void* d_out, int out_size, void* d_ws, size_t ws_size,
                              hipStream_t stream) {
  (void)in_sizes; (void)n_in; (void)out_size; (void)ws_size;
  const int*   y  = (const int*)d_in[0];     // (8,) int32
  const float* yp = (const float*)d_in[1];   // (8,1000) f32
  const float* fm = (const float*)d_in[2];   // (8,64,56,56) f32
  float* out = (float*)d_out;                // scalar f32
  float* ws  = (float*)d_ws;                 // needs ~12.9 MB

  ce_init_kernel<<<1, 256, 0, stream>>>(y, yp, ws);
  relu_boxsum_kernel<<<512, 256, 0, stream>>>(fm, ws + OFF_F, ws + OFF_S);
  pairmax_kernel<<<dim3(10, 64), 256, 0, stream>>>(ws + OFF_F, ws + OFF_S, ws + OFF_M);
  finalize_kernel<<<1, 32, 0, stream>>>(ws, out);
}